// spCRFDecoder_24833500905534
// MI455X (gfx1250) — compile-verified
//
#include <hip/hip_runtime.h>

#define TT   52
#define SOF_T 50
#define EOF_T 51
#define BB   64
#define SS   512
#define DD   1024
#define NPAD 64

typedef __attribute__((ext_vector_type(2))) float v2f;
typedef __attribute__((ext_vector_type(8))) float v8f;

// ---------------------------------------------------------------------------
// Phase A0: zero-pad W (52 x 1024) into Wpad (64 x 1024) so the GEMM's B
// loads need no bounds predication (no exec-mask branches in the hot loop).
// ---------------------------------------------------------------------------
__global__ __launch_bounds__(256) void wpad_kernel(
    const float* __restrict__ W, float* __restrict__ Wpad)
{
    int idx = blockIdx.x * 256 + threadIdx.x;      // 0 .. 64*1024-1
    int n = idx >> 10;
    Wpad[idx] = (n < TT) ? W[idx] : 0.f;
}

// ---------------------------------------------------------------------------
// Phase A: emit[b,s,t] = rep[b,s,:] . W[t,:] + b[t]   via V_WMMA_F32_16X16X4_F32
// One wave computes a 16(M) x 64(N) strip; 8 waves per block -> 128 rows/block.
// All four B fragments come from one base pointer with immediate offsets.
// ---------------------------------------------------------------------------
__global__ __launch_bounds__(256) void emit_gemm_kernel(
    const float* __restrict__ rep, const float* __restrict__ Wpad,
    const float* __restrict__ bias, float* __restrict__ emit)
{
    const int lane  = threadIdx.x & 31;
    const int wave  = threadIdx.x >> 5;
    const int m0    = (blockIdx.x * 8 + wave) * 16;
    const int nA    = lane & 15;
    const int khalf = (lane >> 4) << 1;   // 0 or 2 (K sub-offset per half-wave)

    // A fragment source: row m0+nA, element k+khalf (+1)
    const float* arow = rep  + (size_t)(m0 + nA) * DD + khalf;
    // B fragment base: row nA of Wpad; n-tiles are +16 rows = +16*DD floats
    const float* brow = Wpad + (size_t)nA * DD + khalf;

    v8f acc[4] = {};

    for (int k = 0; k < DD; k += 4) {
        v2f a = *(const v2f*)(arow + k);
        #pragma unroll
        for (int nt = 0; nt < 4; ++nt) {
            v2f bfrag = *(const v2f*)(brow + (size_t)nt * 16 * DD + k);
            acc[nt] = __builtin_amdgcn_wmma_f32_16x16x4_f32(
                false, a, false, bfrag, (short)0, acc[nt], false, false);
        }
    }

    // C layout: VGPR r -> M = m0 + r + 8*(lane>>4), N = nt*16 + (lane&15)
    #pragma unroll
    for (int nt = 0; nt < 4; ++nt) {
        int col = nt * 16 + nA;
        float bv = (col < TT) ? bias[col] : 0.f;
        #pragma unroll
        for (int r = 0; r < 8; ++r) {
            int rowm = m0 + r + ((lane >> 4) << 3);
            emit[(size_t)rowm * NPAD + col] = acc[nt][r] + bv;
        }
    }
}

// ---------------------------------------------------------------------------
// Phase B: CRF forward scan, one 64-thread block per batch element.
// lse(part + trans[:,j]) = m + log( sum_t exp(part_t - m) * E[t][j] ),
// E = exp(transitions) held in 52 registers per thread (constant over steps).
// ---------------------------------------------------------------------------
__global__ __launch_bounds__(64) void crf_scan_kernel(
    const float* __restrict__ emit, const int* __restrict__ lengths,
    const int* __restrict__ tgt, const float* __restrict__ trans,
    float* __restrict__ partials)
{
    const int b = blockIdx.x;
    const int tid = threadIdx.x;
    const bool active = tid < TT;

    __shared__ float p_sh[64];
    __shared__ float tp_sh[64];
    __shared__ float red[2];

    // E column for this thread's state j = tid
    float e[TT];
    #pragma unroll
    for (int t = 0; t < TT; ++t)
        e[t] = active ? __expf(trans[t * TT + tid]) : 0.f;

    const int len = lengths[b];
    const float* erow = emit + (size_t)b * SS * NPAD;

    // partition0 = emit[b,0,:] + transitions[SOF,:]
    float part = active ? (erow[tid] + trans[SOF_T * TT + tid])
                        : -__builtin_huge_valf();

    // target-path energy: sum_{s<len} trans.flat[tgt] + emit[b,s,tgt%T]
    float tp = 0.f;
    for (int s = tid; s < len; s += 64) {
        int tg = tgt[b * SS + s];
        tp += trans[tg] + erow[(size_t)s * NPAD + (tg % TT)];
    }
    tp_sh[tid] = tp;

    // software-pipelined emit row prefetch
    float emit_cur = (len > 1 && active) ? erow[NPAD + tid] : 0.f;

    for (int s = 1; s < len; ++s) {
        // m = max_t part[t]  (wave32 shuffle reduce, combine two waves via LDS)
        float v = active ? part : -__builtin_huge_valf();
        #pragma unroll
        for (int off = 16; off > 0; off >>= 1)
            v = fmaxf(v, __shfl_xor(v, off, 32));
        if ((tid & 31) == 0) red[tid >> 5] = v;
        __syncthreads();
        float m = fmaxf(red[0], red[1]);

        p_sh[tid] = active ? __expf(part - m) : 0.f;

        float emit_next = (s + 1 < len && active)
                            ? erow[(size_t)(s + 1) * NPAD + tid] : 0.f;
        __syncthreads();

        float acc = 0.f;
        #pragma unroll
        for (int t = 0; t < TT; ++t)
            acc = fmaf(p_sh[t], e[t], acc);

        if (active) part = m + __logf(acc) + emit_cur;
        emit_cur = emit_next;
        __syncthreads();   // protect p_sh / red before next iteration writes
    }

    p_sh[tid] = part;
    __syncthreads();
    if (tid == 0) {
        float tps = 0.f;
        for (int i = 0; i < 64; ++i) tps += tp_sh[i];
        partials[b] = p_sh[EOF_T] - tps;
    }
}

// ---------------------------------------------------------------------------
// Phase C: deterministic final reduce of 64 partials -> scalar loss
// ---------------------------------------------------------------------------
__global__ void crf_final_kernel(const float* __restrict__ partials,
                                 float* __restrict__ out)
{
    if (threadIdx.x == 0 && blockIdx.x == 0) {
        float s = 0.f;
        for (int i = 0; i < BB; ++i) s += partials[i];
        out[0] = s / (float)BB;
    }
}

extern "C" void kernel_launch(void* const* d_in, const int* in_sizes, int n_in,
                              void* d_out, int out_size, void* d_ws, size_t ws_size,
                              hipStream_t stream)
{
    const float* rep      = (const float*)d_in[0];
    const int*   lengths  = (const int*)  d_in[1];
    const int*   tgt      = (const int*)  d_in[2];
    const float* W        = (const float*)d_in[3];
    const float* bias     = (const float*)d_in[4];
    const float* trans    = (const float*)d_in[5];

    float* emit     = (float*)d_ws;                       // 32768 x 64 f32 = 8 MB
    float* partials = emit + (size_t)BB * SS * NPAD;      // 64 f32
    float* Wpad     = partials + 64;                      // 64 x 1024 f32 = 256 KB

    wpad_kernel<<<(NPAD * DD) / 256, 256, 0, stream>>>(W, Wpad);
    emit_gemm_kernel<<<(BB * SS) / 128, 256, 0, stream>>>(rep, Wpad, bias, emit);
    crf_scan_kernel<<<BB, 64, 0, stream>>>(emit, lengths, tgt, trans, partials);
    crf_final_kernel<<<1, 32, 0, stream>>>(partials, (float*)d_out);
}